// BFPActivation_57406532878655
// MI455X (gfx1250) — compile-verified
//
#include <hip/hip_runtime.h>
#include <stdint.h>

// BFP (block floating point) quantization, NCHW fp32, block=32 along C.
// Memory-bound: ~205 MB traffic -> ~8.8 us floor @ 23.3 TB/s.
// CDNA5 paths used: global_load_async_to_lds_b64 (ASYNCcnt DMA into LDS),
// s_wait_asynccnt, wave32 tile layout, non-temporal b64 stores.

typedef float f2 __attribute__((ext_vector_type(2)));

#define MANT      5
#define QMAXF     31.0f
#define CDIM      256
#define HW_       3136           // 56*56
#define BLK       32             // channels per BFP block
#define TILE_HW   64             // hw positions per wave tile (49*64 == 3136)
#define NTILES    (HW_ / TILE_HW) // 49, exact
#define WPB       4              // waves per block (128 threads)

__device__ __forceinline__ void bfp_scales(float m, float& inv_s, float& s) {
    // e = floor(log2(m)) for m > 0, denormal-safe, no transcendentals.
    int bits = __float_as_int(m);
    int eb   = (bits >> 23) & 0xff;
    int e    = (eb != 0) ? (eb - 127) : ((31 - __clz(bits)) - 149);
    s     = ldexpf(1.0f, e - (MANT - 1));   // v_ldexp_f32, exact pow2
    inv_s = ldexpf(1.0f, (MANT - 1) - e);   // exact reciprocal pow2
}

__global__ __launch_bounds__(WPB * 32) void bfp_quant_kernel(
    const float* __restrict__ src, float* __restrict__ dst) {
    // 4 waves x (32 rows x 64 floats) = 32 KB LDS per workgroup
    __shared__ f2 tile[WPB][BLK][TILE_HW / 2];

    const int lane = threadIdx.x & 31;
    const int wave = threadIdx.x >> 5;
    const int t    = blockIdx.x * WPB + wave;     // hw tile id (wave-uniform)
    if (t >= NTILES) return;
    const int k = blockIdx.y;                     // channel block id
    const int b = blockIdx.z;                     // batch

    const int    hw0   = t * TILE_HW;
    const size_t chan0 = (size_t)(b * CDIM + k * BLK);

    // ---- async DMA: 32 rows x 256B contiguous -> LDS (one b64 op per row)
    const float* g0 = src + chan0 * HW_ + hw0 + lane * 2;
    uint32_t     l0 = (uint32_t)(uintptr_t)(&tile[wave][0][lane]);
#pragma unroll
    for (int i = 0; i < BLK; ++i) {
        const float* gp = g0 + (size_t)i * HW_;
        uint32_t     lp = l0 + (uint32_t)(i * (TILE_HW / 2) * sizeof(f2));
        asm volatile("global_load_async_to_lds_b64 %0, %1, off"
                     :: "v"(lp), "v"(gp)
                     : "memory");
    }
    asm volatile("s_wait_asynccnt 0" ::: "memory");

    // ---- pass 1: per-lane-pair block maxabs over 32 channels (LDS reads)
    float mx = 0.0f, my = 0.0f;
#pragma unroll
    for (int i = 0; i < BLK; ++i) {
        f2 v = tile[wave][i][lane];
        mx = fmaxf(mx, fabsf(v.x));
        my = fmaxf(my, fabsf(v.y));
    }

    float sx, isx, sy, isy;
    bfp_scales(mx, isx, sx);
    bfp_scales(my, isy, sy);
    const bool zx = (mx == 0.0f);
    const bool zy = (my == 0.0f);

    // ---- pass 2: quantize + non-temporal coalesced stores (256B rows)
    float* o0 = dst + chan0 * HW_ + hw0 + lane * 2;
#pragma unroll
    for (int i = 0; i < BLK; ++i) {
        f2 v = tile[wave][i][lane];
        // round-half-even (v_rndne_f32) matches jnp.round; clamp to +/-31
        float qx = fminf(fmaxf(rintf(v.x * isx), -QMAXF), QMAXF) * sx;
        float qy = fminf(fmaxf(rintf(v.y * isy), -QMAXF), QMAXF) * sy;
        f2 o;
        o.x = zx ? 0.0f : qx;
        o.y = zy ? 0.0f : qy;
        __builtin_nontemporal_store(o, (f2*)(o0 + (size_t)i * HW_));
    }
}

extern "C" void kernel_launch(void* const* d_in, const int* in_sizes, int n_in,
                              void* d_out, int out_size, void* d_ws, size_t ws_size,
                              hipStream_t stream) {
    const float* x = (const float*)d_in[0];
    float*       y = (float*)d_out;

    const int B = in_sizes[0] / (CDIM * HW_);     // 32 for the reference shapes

    dim3 grid((NTILES + WPB - 1) / WPB,           // 13 hw-tile groups
              CDIM / BLK,                         // 8 channel blocks
              B);                                 // batch
    bfp_quant_kernel<<<grid, WPB * 32, 0, stream>>>(x, y);
}